// E2EVIO_478
// MI455X (gfx1250) — compile-verified
//
#include <hip/hip_runtime.h>

typedef float v2f __attribute__((ext_vector_type(2)));
typedef float v8f __attribute__((ext_vector_type(8)));

#define LDW 32

// ---------------- small 3x3 helpers (run scalar on lane 0) ----------------
__device__ __forceinline__ void m3_skew(const float* v, float* K) {
  K[0] = 0.f;   K[1] = -v[2]; K[2] = v[1];
  K[3] = v[2];  K[4] = 0.f;   K[5] = -v[0];
  K[6] = -v[1]; K[7] = v[0];  K[8] = 0.f;
}
__device__ __forceinline__ void m3_mul(const float* A, const float* B, float* C) {
  for (int i = 0; i < 3; ++i)
    for (int j = 0; j < 3; ++j) {
      float s = 0.f;
      for (int k = 0; k < 3; ++k) s += A[i*3+k] * B[k*3+j];
      C[i*3+j] = s;
    }
}
__device__ __forceinline__ void m3_mv(const float* A, const float* x, float* y) {
  for (int i = 0; i < 3; ++i) {
    float s = 0.f;
    for (int k = 0; k < 3; ++k) s += A[i*3+k] * x[k];
    y[i] = s;
  }
}
__device__ __forceinline__ void m3_tr(const float* A, float* At) {
  for (int i = 0; i < 3; ++i)
    for (int j = 0; j < 3; ++j) At[i*3+j] = A[j*3+i];
}
__device__ __forceinline__ void m3_exp(const float* phi, float* R) {
  float th2 = phi[0]*phi[0] + phi[1]*phi[1] + phi[2]*phi[2];
  float th  = sqrtf(fmaxf(th2, 1e-24f));
  float A, Bc;
  if (th < 1e-6f) { A = 1.f - th2/6.f; Bc = 0.5f - th2/24.f; }
  else            { A = sinf(th)/th;   Bc = (1.f - cosf(th))/(th*th); }
  float K[9], K2[9];
  m3_skew(phi, K); m3_mul(K, K, K2);
  for (int i = 0; i < 9; ++i) R[i] = A*K[i] + Bc*K2[i];
  R[0] += 1.f; R[4] += 1.f; R[8] += 1.f;
}
__device__ __forceinline__ void m3_log(const float* C, float* phi) {
  float tr = C[0] + C[4] + C[8];
  float c  = fminf(fmaxf((tr - 1.f)*0.5f, -1.f + 1e-7f), 1.f - 1e-7f);
  float th = acosf(c);
  float f  = (th < 1e-6f) ? (0.5f + th*th/12.f) : (th/(2.f*sinf(th)));
  phi[0] = f*(C[7]-C[5]); phi[1] = f*(C[2]-C[6]); phi[2] = f*(C[3]-C[1]);
}
__device__ __forceinline__ void m3_jlinv(const float* phi, float* J) {
  float th2 = phi[0]*phi[0] + phi[1]*phi[1] + phi[2]*phi[2];
  float th  = sqrtf(fmaxf(th2, 1e-24f));
  float coeff;
  if (th < 1e-6f) coeff = 1.f/12.f + th2/720.f;
  else            coeff = 1.f/(th*th) - (1.f + cosf(th))/(2.f*th*sinf(th));
  float K[9], K2[9];
  m3_skew(phi, K); m3_mul(K, K, K2);
  for (int i = 0; i < 9; ++i) J[i] = -0.5f*K[i] + coeff*K2[i];
  J[0] += 1.f; J[4] += 1.f; J[8] += 1.f;
}

// ---------------- wave-level GEMM on padded 32x32 LDS tiles ----------------
// D[tiles Mt x Nt of 16x16] = A(32xK) * (transB ? B^T : B) (+ Cinit), ld = 32.
// K = 4*Ks.  All buffers zero-padded beyond logical dims so padding is inert.
// Uses V_WMMA_F32_16X16X4_F32: A frag = 16x4 (lanes 0-15: K=0,1; lanes 16-31:
// K=2,3), B frag = 4x16 mirrored, C/D frag = v8f per ISA layout.
__device__ __forceinline__ void wave_gemm(float* __restrict__ D, const float* A,
                                          const float* B, const float* Cinit,
                                          int Mt, int Nt, int Ks, bool transB) {
  const int lane = threadIdx.x & 31;
  const int half = lane >> 4;   // 0 or 1
  const int l16  = lane & 15;
  for (int tm = 0; tm < Mt; ++tm) {
    for (int tn = 0; tn < Nt; ++tn) {
      v8f acc = {0.f, 0.f, 0.f, 0.f, 0.f, 0.f, 0.f, 0.f};
      const int col   = tn*16 + l16;
      const int rbase = tm*16 + half*8;
      if (Cinit) {
#pragma unroll
        for (int r = 0; r < 8; ++r) acc[r] = Cinit[(rbase + r)*LDW + col];
      }
      for (int ks = 0; ks < Ks; ++ks) {
        const int k0 = ks*4;
        v2f a, bf;
        const int arow = tm*16 + l16;
        a.x = A[arow*LDW + k0 + half*2 + 0];
        a.y = A[arow*LDW + k0 + half*2 + 1];
        if (!transB) {
          bf.x = B[(k0 + half*2 + 0)*LDW + col];
          bf.y = B[(k0 + half*2 + 1)*LDW + col];
        } else {
          bf.x = B[col*LDW + (k0 + half*2 + 0)];
          bf.y = B[col*LDW + (k0 + half*2 + 1)];
        }
        acc = __builtin_amdgcn_wmma_f32_16x16x4_f32(false, a, false, bf,
                                                    (short)0, acc, false, false);
      }
#pragma unroll
      for (int r = 0; r < 8; ++r) D[(rbase + r)*LDW + col] = acc[r];
    }
  }
}

__device__ __forceinline__ void zero1024(float* m, int lane) {
  for (int i = lane; i < 1024; i += 32) m[i] = 0.f;
}

// ---------------- main kernel: one wave (32 threads) per batch element -----
// SC scratch layout (floats):
//   0..24  state [g(3) C(9) r(3) v(3) bw(3) ba(3) lam]
//  25..40  pose 4x4
//  41..49  C_acc   50..52 r_acc   53..55 v_acc   56 t_acc
//  57..65  Wk   66..74 Vk   75..83 E   84..92 Et   93..101 CV
// 102..110 Ct  111..119 SG/skew(new_g)
// 120 dt  121 dt2  122..124 a
// 125..127 pred_r  128..130 pred_v  131..133 phi  134..139 res  140..158 err
__global__ __launch_bounds__(32) void ekf_vio_kernel(
    const float* __restrict__ imu,        // (256,64,11,7)
    const float* __restrict__ qn,         // (12,12)
    const float* __restrict__ prev_pose,  // (256,4,4)
    const float* __restrict__ prev_state, // (256,25)
    const float* __restrict__ prev_covar, // (256,19,19)
    const float* __restrict__ vis,        // (256,64,6)
    const float* __restrict__ visc,       // (256,64,6,6)
    float* __restrict__ out_poses,        // (256,65,16)
    float* __restrict__ out_states,       // (256,65,25)
    float* __restrict__ out_covars) {     // (256,65,361)
  const int b    = blockIdx.x;
  const int lane = threadIdx.x;

  __shared__ float lds[9*1024 + 256];
  float* BP  = lds + 0*1024;  // P (covar)
  float* B1  = lds + 1*1024;  // F / H / Uc
  float* B2  = lds + 2*1024;  // Phi / est_covar
  float* BG  = lds + 3*1024;  // G
  float* B4  = lds + 4*1024;  // PhiG / K
  float* BQN = lds + 5*1024;  // Qn (constant)
  float* T6  = lds + 6*1024;  // tmp
  float* T7  = lds + 7*1024;  // tmp2 (vmc/S)
  float* BQ  = lds + 8*1024;  // Q / Sinv
  float* SC  = lds + 9*1024;  // scalar scratch

  // ---- init: zero LDS, load Qn, P, state, pose; emit t=0 outputs ----
  for (int i = lane; i < 9*1024 + 256; i += 32) lds[i] = 0.f;
  for (int i = lane; i < 144; i += 32) BQN[(i/12)*LDW + (i%12)] = qn[i];
  for (int i = lane; i < 361; i += 32) BP[(i/19)*LDW + (i%19)] = prev_covar[b*361 + i];
  if (lane < 25) SC[lane]      = prev_state[b*25 + lane];
  if (lane < 16) SC[25 + lane] = prev_pose[b*16 + lane];
  for (int i = lane; i < 16;  i += 32) out_poses[(b*65 + 0)*16  + i] = prev_pose[b*16 + i];
  for (int i = lane; i < 25;  i += 32) out_states[(b*65 + 0)*25 + i] = prev_state[b*25 + i];
  for (int i = lane; i < 361; i += 32) out_covars[(b*65 + 0)*361 + i] = prev_covar[b*361 + i];

  for (int t = 0; t < 64; ++t) {
    // ---- P0 = covar * Umask (zero rows/cols 3..8) ----
    for (int idx = lane; idx < 361; idx += 32) {
      int i = idx / 19, j = idx % 19;
      if ((i >= 3 && i < 9) || (j >= 3 && j < 9)) BP[i*LDW + j] = 0.f;
    }
    if (lane == 0) {
      for (int i = 0; i < 9; ++i) SC[41 + i] = 0.f;
      SC[41] = 1.f; SC[45] = 1.f; SC[49] = 1.f;     // C_acc = I
      for (int i = 0; i < 3; ++i) { SC[50 + i] = 0.f; SC[53 + i] = 0.f; }
      SC[56] = 0.f;                                  // t_acc = 0
    }

    // ================= predict: 10 IMU substeps =================
    for (int n = 0; n < 10; ++n) {
      const long base = ((long)(b*64 + t)*11 + n)*7;
      if (lane == 0) {
        float dt = imu[base + 7] - imu[base + 0];
        SC[120] = dt; SC[121] = dt*dt;
        float w[3];
        for (int i = 0; i < 3; ++i) {
          w[i]        = imu[base + 1 + i] - SC[18 + i];
          SC[122 + i] = imu[base + 4 + i] - SC[21 + i];  // a
        }
        m3_skew(w, SC + 57);                 // Wk
        m3_tr(SC + 41, SC + 102);            // Ct = C_acc^T
        float ta = SC[56], tv[3], vloc[3];
        for (int i = 0; i < 3; ++i) tv[i] = SC[15 + i] - SC[i]*ta + SC[53 + i];
        m3_mv(SC + 102, tv, vloc);
        m3_skew(vloc, SC + 66);              // Vk
        float dphi[3] = {dt*w[0], dt*w[1], dt*w[2]};
        m3_exp(dphi, SC + 75);               // E
        m3_tr(SC + 75, SC + 84);             // Et
        m3_mul(SC + 41, SC + 66, SC + 93);   // CV = C_acc @ Vk
        float ctg[3]; m3_mv(SC + 102, SC, ctg);
        m3_skew(ctg, SC + 111);              // SG = skew(Ct@g)
      }

      // ---- build F ----
      zero1024(B1, lane);
      if (lane == 0) {
        for (int r = 0; r < 3; ++r)
          for (int c = 0; c < 3; ++c) {
            float I = (r == c) ? 1.f : 0.f;
            B1[(3+r)*LDW + (3+c)]  = -SC[57 + r*3 + c];
            B1[(3+r)*LDW + (12+c)] = -I;
            B1[(6+r)*LDW + (3+c)]  = -SC[93 + r*3 + c];
            B1[(6+r)*LDW + (9+c)]  =  SC[41 + r*3 + c];
            B1[(9+r)*LDW + (c)]    = -SC[102 + r*3 + c];
            B1[(9+r)*LDW + (3+c)]  = -SC[111 + r*3 + c];
            B1[(9+r)*LDW + (9+c)]  = -SC[57 + r*3 + c];
            B1[(9+r)*LDW + (12+c)] = -SC[66 + r*3 + c];
            B1[(9+r)*LDW + (15+c)] = -I;
          }
      }
      wave_gemm(T6, B1, B1, nullptr, 2, 2, 5, false);   // FF = F@F

      // ---- Phi = I + F dt + 0.5 FF dt^2 ; patch blocks ----
      {
        float dt = SC[120], dt2 = SC[121];
        for (int i = lane; i < 1024; i += 32) {
          int rr = i >> 5, cc = i & 31;
          float id = ((rr == cc) && (rr < 19)) ? 1.f : 0.f;
          B2[i] = id + B1[i]*dt + 0.5f*T6[i]*dt2;
        }
      }
      if (lane == 0) {
        for (int r = 0; r < 3; ++r)
          for (int c = 0; c < 3; ++c) {
            B2[(6+r)*LDW + (12+c)] = 0.f;
            B2[(3+r)*LDW + (3+c)]  = SC[84 + r*3 + c];
            B2[(9+r)*LDW + (9+c)]  = SC[84 + r*3 + c];
          }
        // ---- build G (fixed sparsity; buffer otherwise stays zero) ----
        for (int r = 0; r < 3; ++r)
          for (int c = 0; c < 3; ++c) {
            float I = (r == c) ? 1.f : 0.f;
            BG[(3+r)*LDW + (c)]    = -I;
            BG[(9+r)*LDW + (c)]    = -SC[66 + r*3 + c];
            BG[(9+r)*LDW + (6+c)]  = -I;
            BG[(12+r)*LDW + (3+c)] =  I;
            BG[(15+r)*LDW + (9+c)] =  I;
          }
      }

      wave_gemm(B4, B2, BG, nullptr, 2, 1, 5, false);   // PhiG (19x12)
      wave_gemm(T6, B4, BQN, nullptr, 2, 1, 3, false);  // PhiG @ Qn
      wave_gemm(BQ, T6, B4, nullptr, 2, 2, 3, true);    // Q0 = (PhiG Qn) PhiG^T
      {
        float dt = SC[120];
        for (int i = lane; i < 1024; i += 32) BQ[i] *= dt;   // Q = Q0 * dt
      }
      wave_gemm(T7, B2, BP, nullptr, 2, 2, 5, false);   // Phi @ P
      wave_gemm(BP, T7, B2, BQ, 2, 2, 5, true);         // P = (PhiP)Phi^T + Q
      for (int idx = lane; idx < 361; idx += 32) {      // force_sym
        int i = idx / 19, j = idx % 19;
        if (i < j) BP[j*LDW + i] = BP[i*LDW + j];
      }

      // ---- accumulator updates ----
      if (lane == 0) {
        float dt = SC[120], dt2 = SC[121], tmp[3], o[3];
        for (int i = 0; i < 3; ++i) tmp[i] = dt2*SC[122 + i];
        m3_mv(SC + 41, tmp, o);
        for (int i = 0; i < 3; ++i) SC[50 + i] += SC[53 + i]*dt + 0.5f*o[i];
        for (int i = 0; i < 3; ++i) tmp[i] = dt*SC[122 + i];
        m3_mv(SC + 41, tmp, o);
        for (int i = 0; i < 3; ++i) SC[53 + i] += o[i];
        float nc[9]; m3_mul(SC + 41, SC + 75, nc);
        for (int i = 0; i < 9; ++i) SC[41 + i] = nc[i];
        SC[56] += dt;
      }
    }

    // ================= measurement update =================
    const float* vm = vis + (long)(b*64 + t)*6;
    if (lane == 0) {
      float ta = SC[56], pr[3], pv[3], ph[3], CAt[9], tv[3];
      for (int i = 0; i < 3; ++i) pr[i] = SC[15 + i]*ta - 0.5f*SC[i]*ta*ta + SC[50 + i];
      m3_tr(SC + 41, CAt);
      for (int i = 0; i < 3; ++i) tv[i] = SC[15 + i] - SC[i]*ta + SC[53 + i];
      m3_mv(CAt, tv, pv);
      m3_log(SC + 41, ph);
      for (int i = 0; i < 3; ++i) {
        SC[125 + i] = pr[i]; SC[128 + i] = pv[i]; SC[131 + i] = ph[i];
        SC[134 + i] = vm[i] - ph[i];
        SC[137 + i] = vm[3 + i] - SC[24]*pr[i];
      }
    }
    // ---- H (after H = -H folding): H[0:3,3:6]=Jl_inv(-phi); H[3:6,6:9]=lam I;
    //      H[3:6,18]=pred_r ----
    zero1024(B1, lane);
    if (lane == 0) {
      float nph[3] = {-SC[131], -SC[132], -SC[133]};
      float J[9]; m3_jlinv(nph, J);
      for (int r = 0; r < 3; ++r)
        for (int c = 0; c < 3; ++c) B1[r*LDW + 3 + c] = J[r*3 + c];
      for (int r = 0; r < 3; ++r) {
        B1[(3+r)*LDW + (6+r)] = SC[24];
        B1[(3+r)*LDW + 18]    = SC[125 + r];
      }
    }
    // vmc into T7
    zero1024(T7, lane);
    for (int i = lane; i < 36; i += 32)
      T7[(i/6)*LDW + (i%6)] = visc[(long)(b*64 + t)*36 + i];

    wave_gemm(T6, B1, BP, nullptr, 1, 2, 5, false);   // HP = H @ P
    wave_gemm(T7, T6, B1, T7, 1, 1, 5, true);         // S = HP H^T + vmc

    // ---- Sinv via Gauss-Jordan (lane 0, in LDS; result padded into BQ) ----
    zero1024(BQ, lane);
    if (lane == 0) {
      for (int r = 0; r < 6; ++r)
        for (int c = 0; c < 6; ++c) {
          BQ[r*LDW + c]     = T7[r*LDW + c];
          BQ[r*LDW + 6 + c] = (r == c) ? 1.f : 0.f;
        }
      for (int col = 0; col < 6; ++col) {
        int p = col; float best = fabsf(BQ[col*LDW + col]);
        for (int r = col + 1; r < 6; ++r) {
          float v = fabsf(BQ[r*LDW + col]);
          if (v > best) { best = v; p = r; }
        }
        if (p != col)
          for (int c = 0; c < 12; ++c) {
            float tv = BQ[col*LDW + c]; BQ[col*LDW + c] = BQ[p*LDW + c]; BQ[p*LDW + c] = tv;
          }
        float inv = 1.f / BQ[col*LDW + col];
        for (int c = 0; c < 12; ++c) BQ[col*LDW + c] *= inv;
        for (int r = 0; r < 6; ++r)
          if (r != col) {
            float f = BQ[r*LDW + col];
            for (int c = 0; c < 12; ++c) BQ[r*LDW + c] -= f*BQ[col*LDW + c];
          }
      }
      for (int r = 0; r < 6; ++r)
        for (int c = 0; c < 6; ++c) {
          BQ[r*LDW + c] = BQ[r*LDW + 6 + c];
          BQ[r*LDW + 6 + c] = 0.f;
        }
    }

    wave_gemm(T6, BP, B1, nullptr, 2, 1, 5, true);    // P H^T (19x6)
    wave_gemm(B4, T6, BQ, nullptr, 2, 1, 2, false);   // K = (P H^T) Sinv
    wave_gemm(T6, B4, B1, nullptr, 2, 2, 2, false);   // K H
    for (int i = lane; i < 1024; i += 32) {           // I - KH
      int rr = i >> 5, cc = i & 31;
      float id = ((rr == cc) && (rr < 19)) ? 1.f : 0.f;
      T6[i] = id - T6[i];
    }
    wave_gemm(B2, T6, BP, nullptr, 2, 2, 5, false);   // est = (I-KH) P

    // ---- err = K @ res ----
    if (lane < 19) {
      float s = 0.f;
      for (int j = 0; j < 6; ++j) s += B4[lane*LDW + j]*SC[134 + j];
      SC[140 + lane] = s;
    }

    // ---- state / pose update ----
    if (lane == 0) {
      float* err = SC + 140;
      float dphi[3] = {err[3], err[4], err[5]};
      float Eu[9], Ce[9], ge[3], re[3], ve[3], Ct[9];
      m3_exp(dphi, Eu);
      m3_mul(SC + 41, Eu, Ce);                  // C_e = pred_C @ exp(err)
      for (int i = 0; i < 3; ++i) {
        ge[i] = SC[i] + err[i];
        re[i] = SC[125 + i] + err[6 + i];
        ve[i] = SC[128 + i] + err[9 + i];
        SC[18 + i] += err[12 + i];
        SC[21 + i] += err[15 + i];
      }
      SC[24] += err[18];
      m3_tr(Ce, Ct);
      float R0[9], t0[3], Rn[9], d3[3], tn[3];
      for (int r = 0; r < 3; ++r) {
        for (int c = 0; c < 3; ++c) R0[r*3 + c] = SC[25 + r*4 + c];
        t0[r] = SC[25 + r*4 + 3];
      }
      m3_mul(Ct, R0, Rn);
      for (int i = 0; i < 3; ++i) d3[i] = t0[i] - re[i];
      m3_mv(Ct, d3, tn);
      for (int r = 0; r < 3; ++r) {
        for (int c = 0; c < 3; ++c) SC[25 + r*4 + c] = Rn[r*3 + c];
        SC[25 + r*4 + 3] = tn[r];
      }
      SC[25 + 12] = 0.f; SC[25 + 13] = 0.f; SC[25 + 14] = 0.f; SC[25 + 15] = 1.f;
      float ng[3]; m3_mv(Ct, ge, ng);
      for (int i = 0; i < 3; ++i) { SC[i] = ng[i]; SC[12 + i] = re[i]; SC[15 + i] = ve[i]; }
      for (int i = 0; i < 9; ++i) { SC[3 + i] = Ce[i]; SC[102 + i] = Ct[i]; }
      m3_skew(ng, SC + 111);
    }

    // ---- Uc; new_covar = force_sym(Uc est Uc^T) ----
    zero1024(B1, lane);
    if (lane < 19) B1[lane*LDW + lane] = 1.f;
    if (lane == 0) {
      for (int r = 0; r < 3; ++r)
        for (int c = 0; c < 3; ++c) {
          B1[r*LDW + c]     = SC[102 + r*3 + c];   // Ct
          B1[r*LDW + 3 + c] = SC[111 + r*3 + c];   // skew(new_g)
        }
    }
    wave_gemm(T6, B1, B2, nullptr, 2, 2, 5, false);   // Uc @ est
    wave_gemm(BP, T6, B1, nullptr, 2, 2, 5, true);    // @ Uc^T
    for (int idx = lane; idx < 361; idx += 32) {      // force_sym
      int i = idx / 19, j = idx % 19;
      if (i < j) BP[j*LDW + i] = BP[i*LDW + j];
    }

    // ---- emit outputs at t+1 ----
    const long o = (long)b*65 + (t + 1);
    for (int i = lane; i < 16;  i += 32) out_poses[o*16 + i]  = SC[25 + i];
    for (int i = lane; i < 25;  i += 32) out_states[o*25 + i] = SC[i];
    for (int i = lane; i < 361; i += 32) out_covars[o*361 + i] = BP[(i/19)*LDW + (i%19)];
  }
}

extern "C" void kernel_launch(void* const* d_in, const int* in_sizes, int n_in,
                              void* d_out, int out_size, void* d_ws, size_t ws_size,
                              hipStream_t stream) {
  const float* imu    = (const float*)d_in[0];
  const float* qn     = (const float*)d_in[1];
  const float* ppose  = (const float*)d_in[2];
  const float* pstate = (const float*)d_in[3];
  const float* pcov   = (const float*)d_in[4];
  const float* vis    = (const float*)d_in[5];
  const float* visc   = (const float*)d_in[6];
  float* out        = (float*)d_out;
  float* out_poses  = out;
  float* out_states = out + (long)256*65*16;
  float* out_covars = out + (long)256*65*16 + (long)256*65*25;
  ekf_vio_kernel<<<256, 32, 0, stream>>>(imu, qn, ppose, pstate, pcov, vis, visc,
                                         out_poses, out_states, out_covars);
}